// PointCloudVoxelizer_62388694942111
// MI455X (gfx1250) — compile-verified
//
#include <hip/hip_runtime.h>

#define G 64
#define BATCH 16
#define EPSV 1e-6f
#define FLT_MAXV 3.402823466e+38f

// Order-preserving encode: float total order == unsigned total order.
__device__ __forceinline__ unsigned encodeF(float f) {
    unsigned u = __float_as_uint(f);
    return (u & 0x80000000u) ? ~u : (u | 0x80000000u);
}
__device__ __forceinline__ float decodeF(unsigned u) {
    unsigned v = (u & 0x80000000u) ? (u & 0x7FFFFFFFu) : ~u;
    return __uint_as_float(v);
}

// ws layout: [0 .. 3*BATCH)       encoded per-batch mins (init 0xFFFFFFFF)
//            [3*BATCH .. 6*BATCH) encoded per-batch maxs (init 0)

// ---------------------------------------------------------------------------
// Histogram kernel FIRST in the file so its disassembly lands in the snippet
// window — verifying the fp32 atomic lowers to global_atomic_add_f32.
// ---------------------------------------------------------------------------
__global__ __launch_bounds__(256) void voxel_hist_kernel(const float* __restrict__ pts,
                                                         const unsigned* __restrict__ ws,
                                                         float* __restrict__ out,
                                                         int n) {
#pragma clang fp contract(off)
    const int b = blockIdx.y;

    float pmin0 = decodeF(ws[b * 3 + 0]);
    float pmin1 = decodeF(ws[b * 3 + 1]);
    float pmin2 = decodeF(ws[b * 3 + 2]);
    float den0 = (decodeF(ws[3 * BATCH + b * 3 + 0]) - pmin0) + EPSV;
    float den1 = (decodeF(ws[3 * BATCH + b * 3 + 1]) - pmin1) + EPSV;
    float den2 = (decodeF(ws[3 * BATCH + b * 3 + 2]) - pmin2) + EPSV;

    const float* __restrict__ p = pts + (size_t)b * (size_t)n * 3u;
    float* __restrict__ outb = out + (size_t)b * (size_t)(G * G * G);

    const int stride = gridDim.x * blockDim.x;
    for (int i = blockIdx.x * blockDim.x + threadIdx.x; i < n; i += stride) {
        const float* q = p + (size_t)i * 3u;
        // gfx1250 global_prefetch_b8 for the next grid-stride iteration
        __builtin_prefetch(q + (size_t)stride * 3u, 0, 0);

        float x = q[0], y = q[1], z = q[2];

        // Replicate reference arithmetic exactly (no FMA contraction):
        // norm = (p - pmin) / (pmax - pmin + EPS) * 2 - 1
        float nx = (x - pmin0) / den0 * 2.0f - 1.0f;
        float ny = (y - pmin1) / den1 * 2.0f - 1.0f;
        float nz = (z - pmin2) / den2 * 2.0f - 1.0f;

        bool ok = (nx >= -1.0f) & (nx <= 1.0f) &
                  (ny >= -1.0f) & (ny <= 1.0f) &
                  (nz >= -1.0f) & (nz <= 1.0f);

        // idx = clip(floor((norm + 1) * 0.5 * G), 0, G-1); 0.5*G == 32
        int ix = (int)floorf((nx + 1.0f) * 32.0f);
        int iy = (int)floorf((ny + 1.0f) * 32.0f);
        int iz = (int)floorf((nz + 1.0f) * 32.0f);
        ix = min(max(ix, 0), G - 1);
        iy = min(max(iy, 0), G - 1);
        iz = min(max(iz, 0), G - 1);

        if (ok) {
            int flat = (ix * G + iy) * G + iz;
            // expect: global_atomic_add_f32 — histogram (16 MB) is L2-resident
            unsafeAtomicAdd(&outb[flat], 1.0f);
        }
    }
}

__global__ __launch_bounds__(256) void voxel_minmax_kernel(const float* __restrict__ pts,
                                                           unsigned* __restrict__ ws,
                                                           int n) {
    __shared__ unsigned smin[3];
    __shared__ unsigned smax[3];
    if (threadIdx.x < 3) { smin[threadIdx.x] = 0xFFFFFFFFu; smax[threadIdx.x] = 0u; }
    __syncthreads();

    const int b = blockIdx.y;
    const float* __restrict__ p = pts + (size_t)b * (size_t)n * 3u;

    float mn0 = FLT_MAXV, mn1 = FLT_MAXV, mn2 = FLT_MAXV;
    float mx0 = -FLT_MAXV, mx1 = -FLT_MAXV, mx2 = -FLT_MAXV;

    const int stride = gridDim.x * blockDim.x;
    for (int i = blockIdx.x * blockDim.x + threadIdx.x; i < n; i += stride) {
        const float* q = p + (size_t)i * 3u;
        float x = q[0], y = q[1], z = q[2];
        mn0 = fminf(mn0, x); mx0 = fmaxf(mx0, x);
        mn1 = fminf(mn1, y); mx1 = fmaxf(mx1, y);
        mn2 = fminf(mn2, z); mx2 = fmaxf(mx2, z);
    }

    // LDS reduction via ds_min_u32 / ds_max_u32 on order-preserving encodings.
    atomicMin(&smin[0], encodeF(mn0));
    atomicMin(&smin[1], encodeF(mn1));
    atomicMin(&smin[2], encodeF(mn2));
    atomicMax(&smax[0], encodeF(mx0));
    atomicMax(&smax[1], encodeF(mx1));
    atomicMax(&smax[2], encodeF(mx2));
    __syncthreads();

    if (threadIdx.x < 3) {
        atomicMin(&ws[b * 3 + threadIdx.x], smin[threadIdx.x]);
        atomicMax(&ws[3 * BATCH + b * 3 + threadIdx.x], smax[threadIdx.x]);
    }
}

// One element per thread: grid sized exactly at launch, no divide in codegen.
__global__ __launch_bounds__(256) void voxel_init_kernel(float* __restrict__ out,
                                                         int out_n,
                                                         unsigned* __restrict__ ws) {
    int i = blockIdx.x * blockDim.x + threadIdx.x;
    if (i < out_n) out[i] = 0.0f;
    if (i < 3 * BATCH)      ws[i] = 0xFFFFFFFFu;  // min slots
    else if (i < 6 * BATCH) ws[i] = 0u;           // max slots
}

extern "C" void kernel_launch(void* const* d_in, const int* in_sizes, int n_in,
                              void* d_out, int out_size, void* d_ws, size_t ws_size,
                              hipStream_t stream) {
    const float* pts = (const float*)d_in[0];
    float* out = (float*)d_out;
    unsigned* ws = (unsigned*)d_ws;

    const int n = in_sizes[0] / (BATCH * 3);  // points per batch (500000)

    // 1) zero output histogram + seed encoded min/max slots
    int init_blocks = (out_size + 255) / 256;
    voxel_init_kernel<<<init_blocks, 256, 0, stream>>>(out, out_size, ws);

    // 2) per-batch min/max reduction
    dim3 gmm(128, BATCH);
    voxel_minmax_kernel<<<gmm, 256, 0, stream>>>(pts, ws, n);

    // 3) binning + L2-resident float atomic histogram
    dim3 gh(256, BATCH);
    voxel_hist_kernel<<<gh, 256, 0, stream>>>(pts, ws, out, n);
}